// StandAloneAttention_26156350832782
// MI455X (gfx1250) — compile-verified
//
#include <hip/hip_runtime.h>
#include <hip/hip_bf16.h>

// Problem constants (from reference)
#define Bc 4
#define Cc 64
#define Hc 128
#define Wc 128
#define KHc 5
#define KWc 5
#define PADc 2

#define TILE_H 8
#define LDS_H (TILE_H + 2 * PADc)   // 12
#define LDS_W (Wc + 2 * PADc)       // 132
#define NTHREADS 256

// async-to-LDS builtin expects addrspace(1) int* (global) and addrspace(3) int* (LDS)
typedef __attribute__((address_space(1))) int* gas_ptr;
typedef __attribute__((address_space(3))) int* lds_ptr;

__global__ __launch_bounds__(NTHREADS)
void saa_kernel(const float* __restrict__ f1,
                const float* __restrict__ f2,
                const float* __restrict__ relh_g,
                const float* __restrict__ relw_g,
                float* __restrict__ out) {
    __shared__ float tile[LDS_H * LDS_W];

    const int bi   = blockIdx.x;
    const int tIdx = bi & 15;            // H / TILE_H = 16 strips
    const int cin  = (bi >> 4) & 63;     // input channel
    const int b    = bi >> 10;
    const int y0   = tIdx * TILE_H;
    const int tid  = threadIdx.x;

    const float* __restrict__ plane = f2 + (size_t)((b * Cc + cin) * Hc) * Wc;

    // ---- 1) zero-fill LDS tile (provides the PAD=2 zero border) ----
    for (int i = tid; i < LDS_H * LDS_W; i += NTHREADS) tile[i] = 0.0f;
    __syncthreads();

    // ---- 2) async Global -> LDS fill of in-range elements (CDNA5 async engine) ----
    for (int i = tid; i < LDS_H * LDS_W; i += NTHREADS) {
        const int r  = i / LDS_W;
        const int cc = i - r * LDS_W;
        const int gy = y0 + r - PADc;
        const int gx = cc - PADc;
        if (gy >= 0 && gy < Hc && gx >= 0 && gx < Wc) {
            const float* src = plane + gy * Wc + gx;
#if __has_builtin(__builtin_amdgcn_global_load_async_to_lds_b32)
            __builtin_amdgcn_global_load_async_to_lds_b32(
                (gas_ptr)(unsigned long long)src,
                (lds_ptr)(unsigned int)(unsigned long long)(&tile[i]),
                0 /*offset*/, 0 /*cpol*/);
#else
            tile[i] = *src;
#endif
        }
    }
#if __has_builtin(__builtin_amdgcn_s_wait_asynccnt)
    __builtin_amdgcn_s_wait_asynccnt(0);
#else
    asm volatile("s_wait_asynccnt 0x0" ::: "memory");
#endif
    __syncthreads();

    // ---- 3) per-channel positional bias (5 uniform scalars) ----
    const bool useH = (cin < (Cc / 2));
    float relv[5];
    if (useH) {
#pragma unroll
        for (int t = 0; t < KHc; ++t) relv[t] = relh_g[cin * KHc + t];
    } else {
#pragma unroll
        for (int t = 0; t < KWc; ++t) relv[t] = relw_g[(cin - Cc / 2) * KWc + t];
    }

    // output channel permutation: c_in = head*8 + dh  ->  c_out = dh*8 + head
    const int cout = ((cin & 7) << 3) | (cin >> 3);

    const float* __restrict__ qplane = f1  + (size_t)((b * Cc + cin)  * Hc + y0) * Wc;
    float*       __restrict__ oplane = out + (size_t)((b * Cc + cout) * Hc + y0) * Wc;

    // ---- 4) per-pixel windowed softmax-attention (4 pixels / thread) ----
#pragma unroll
    for (int it = 0; it < (TILE_H * Wc) / NTHREADS; ++it) {
        const int p  = tid + it * NTHREADS;
        const int py = p >> 7;        // / 128
        const int px = p & 127;      // % 128

        const float q = qplane[py * Wc + px];

        float wv[KHc * KWc];
        float m = -3.402823466e+38f;
#pragma unroll
        for (int kh = 0; kh < KHc; ++kh) {
#pragma unroll
            for (int kw = 0; kw < KWc; ++kw) {
                const float v = tile[(py + kh) * LDS_W + (px + kw)];
                wv[kh * KWc + kw] = v;
                const float rb = useH ? relv[kh] : relv[kw];
                const float s  = q * (v + rb);
                m = fmaxf(m, s);
            }
        }

        float denom = 0.0f, num = 0.0f;
#pragma unroll
        for (int kh = 0; kh < KHc; ++kh) {
#pragma unroll
            for (int kw = 0; kw < KWc; ++kw) {
                const float v  = wv[kh * KWc + kw];
                const float rb = useH ? relv[kh] : relv[kw];
                const float e  = __expf(q * (v + rb) - m);   // v_exp_f32
                denom += e;
                num   += e * v;
            }
        }
        oplane[py * Wc + px] = num / denom;
    }
}

extern "C" void kernel_launch(void* const* d_in, const int* in_sizes, int n_in,
                              void* d_out, int out_size, void* d_ws, size_t ws_size,
                              hipStream_t stream) {
    const float* f1 = (const float*)d_in[0];   // features1 [B,C,H,W]
    const float* f2 = (const float*)d_in[1];   // features2 [B,C,H,W]
    const float* rh = (const float*)d_in[2];   // rel_h     [C/2,1,1,KH,1]
    const float* rw = (const float*)d_in[3];   // rel_w     [C/2,1,1,1,KW]
    float* out = (float*)d_out;                // [B,C,H,W]

    const int nblocks = Bc * Cc * (Hc / TILE_H);   // 4096
    saa_kernel<<<nblocks, NTHREADS, 0, stream>>>(f1, f2, rh, rw, out);
}